// QuantumFeatureExtractor_90280212562114
// MI455X (gfx1250) — compile-verified
//
#include <hip/hip_runtime.h>

typedef __attribute__((ext_vector_type(2))) float v2f;
typedef __attribute__((ext_vector_type(8))) float v8f;

#define N_LAYERS 3
#define TILES_PER_WAVE 4
#define WAVES_PER_BLOCK 8

struct cplx { float re, im; };

__device__ __forceinline__ cplx cmul(cplx a, cplx b) {
    cplx r; r.re = a.re*b.re - a.im*b.im; r.im = a.re*b.im + a.im*b.re; return r;
}

// Half-wave swap (lane ^= 16) via ds_swizzle immediate: group-of-32 mode,
// and_mask=0x1f, or_mask=0, xor_mask=0x10 -> offset = (0x10<<10)|0x1f = 0x401f
__device__ __forceinline__ float xor16(float v) {
    return __int_as_float(__builtin_amdgcn_ds_swizzle(__float_as_int(v), 0x401f));
}

__device__ __forceinline__ v2f mkv2(float a, float b) {
    v2f v; v[0] = a; v[1] = b; return v;
}

__device__ __forceinline__ v8f wmma_f32_k4(v2f a, v2f b, v8f c) {
    // D(16x16,f32) = A(16x4,f32) x B(4x16,f32) + C
    return __builtin_amdgcn_wmma_f32_16x16x4_f32(
        false, a, false, b, (short)0, c, false, false);
}

// RX gate on an in-lane row bit S: RX = [[c, -i s],[-i s, c]]
template<int S>
__device__ __forceinline__ void rx_inlane(float (&sr)[8], float (&si)[8], float c, float s) {
#pragma unroll
    for (int r0 = 0; r0 < 8; ++r0) {
        if (r0 & S) continue;
        const int r1 = r0 | S;
        const float a0r = sr[r0], a0i = si[r0], a1r = sr[r1], a1i = si[r1];
        sr[r0] = __builtin_fmaf(c, a0r,  s * a1i);
        si[r0] = __builtin_fmaf(c, a0i, -s * a1r);
        sr[r1] = __builtin_fmaf(c, a1r,  s * a0i);
        si[r1] = __builtin_fmaf(c, a1i, -s * a0r);
    }
}

// RX embedding for one layer on a live state:
//  - qubits 1,2,3 are in-lane row bits 2,1,0
//  - qubit 0 is the half-wave bit: one half-swap provides BOTH the gate input
//    and (fused) the swapped post-gate state needed for the WMMA B layout.
__device__ __forceinline__ void rx_layer_live(
    float (&sr)[8], float (&si)[8], float (&trr)[8], float (&tri)[8],
    float c0, float s0, float c1, float s1, float c2, float s2, float c3, float s3)
{
    rx_inlane<4>(sr, si, c1, s1);
    rx_inlane<2>(sr, si, c2, s2);
    rx_inlane<1>(sr, si, c3, s3);
#pragma unroll
    for (int r = 0; r < 8; ++r) {
        const float pr = xor16(sr[r]);
        const float pi = xor16(si[r]);
        const float ar = sr[r], ai = si[r];
        sr[r]  = __builtin_fmaf(c0, ar,  s0 * pi);   // own post-RX
        si[r]  = __builtin_fmaf(c0, ai, -s0 * pr);
        trr[r] = __builtin_fmaf(c0, pr,  s0 * ai);   // partner post-RX (no 2nd shuffle)
        tri[r] = __builtin_fmaf(c0, pi, -s0 * ar);
    }
}

// One layer of the fixed 16x16 complex unitary (Rot (x)4, fused CNOT-chain row
// permutation) applied to 16 batch columns via f32 WMMA. trr/tri is the
// half-swapped state (rows of the other half-wave).
__device__ __forceinline__ void layer_matmul(
    const float (&Ar)[4][2], const float (&Ai)[4][2],
    float (&sr)[8], float (&si)[8],
    const float (&trr)[8], const float (&tri)[8], bool lo)
{
    v2f Br[4], Bi[4], Bn[4];
    Br[0] = mkv2(lo?sr[0]:trr[2], lo?sr[1]:trr[3]);
    Br[1] = mkv2(lo?sr[4]:trr[6], lo?sr[5]:trr[7]);
    Br[2] = mkv2(lo?trr[0]:sr[2], lo?trr[1]:sr[3]);
    Br[3] = mkv2(lo?trr[4]:sr[6], lo?trr[5]:sr[7]);
    Bi[0] = mkv2(lo?si[0]:tri[2], lo?si[1]:tri[3]);
    Bi[1] = mkv2(lo?si[4]:tri[6], lo?si[5]:tri[7]);
    Bi[2] = mkv2(lo?tri[0]:si[2], lo?tri[1]:si[3]);
    Bi[3] = mkv2(lo?tri[4]:si[6], lo?tri[5]:si[7]);
#pragma unroll
    for (int k = 0; k < 4; ++k) Bn[k] = mkv2(-Bi[k][0], -Bi[k][1]);

    v8f Dr, Di;
#pragma unroll
    for (int r = 0; r < 8; ++r) { Dr[r] = 0.0f; Di[r] = 0.0f; }
#pragma unroll
    for (int k = 0; k < 4; ++k) {
        const v2f Arf = mkv2(Ar[k][0], Ar[k][1]);
        const v2f Aif = mkv2(Ai[k][0], Ai[k][1]);
        Dr = wmma_f32_k4(Arf, Br[k], Dr);   // + Ar*Br
        Dr = wmma_f32_k4(Aif, Bn[k], Dr);   // - Ai*Bi
        Di = wmma_f32_k4(Arf, Bi[k], Di);   // + Ar*Bi
        Di = wmma_f32_k4(Aif, Br[k], Di);   // + Ai*Br
    }
#pragma unroll
    for (int r = 0; r < 8; ++r) { sr[r] = Dr[r]; si[r] = Di[r]; }
}

__global__ __launch_bounds__(256) void qfe_wmma_kernel(
    const float* __restrict__ x, const float* __restrict__ w,
    float* __restrict__ out, int n_tiles)
{
    const int lane = (int)(threadIdx.x & 31u);
    const int m    = lane & 15;        // A-frag row / B-frag column
    const int sub  = lane >> 4;        // half-wave id
    const bool lo  = (sub == 0);

    // Per-layer fixed operator M = P_cnot * (Rot0 (x) Rot1 (x) Rot2 (x) Rot3)
    // as f32 WMMA A-fragments; inverse CNOT-chain row permutation = Gray code.
    const int gi = m ^ (m >> 1);
    float ArF[N_LAYERS][4][2], AiF[N_LAYERS][4][2];
#pragma unroll
    for (int l = 0; l < N_LAYERS; ++l) {
        cplx R[4][2][2];
#pragma unroll
        for (int q = 0; q < 4; ++q) {
            const float phi = w[(l*4 + q)*3 + 0];
            const float th  = w[(l*4 + q)*3 + 1];
            const float om  = w[(l*4 + q)*3 + 2];
            float st, ct, sa, ca, sb, cb;
            __sincosf(0.5f*th,       &st, &ct);
            __sincosf(0.5f*(phi+om), &sa, &ca);
            __sincosf(0.5f*(phi-om), &sb, &cb);
            R[q][0][0].re =  ct*ca;  R[q][0][0].im = -ct*sa;
            R[q][0][1].re = -st*cb;  R[q][0][1].im = -st*sb;
            R[q][1][0].re =  st*cb;  R[q][1][0].im = -st*sb;
            R[q][1][1].re =  ct*ca;  R[q][1][1].im =  ct*sa;
        }
#pragma unroll
        for (int k = 0; k < 4; ++k) {
#pragma unroll
            for (int t = 0; t < 2; ++t) {
                const int j = 4*k + 2*sub + t;
                cplx e =    R[0][(gi>>3)&1][(j>>3)&1];
                e = cmul(e, R[1][(gi>>2)&1][(j>>2)&1]);
                e = cmul(e, R[2][(gi>>1)&1][(j>>1)&1]);
                e = cmul(e, R[3][ gi     &1][ j     &1]);
                ArF[l][k][t] = e.re;
                AiF[l][k][t] = e.im;
            }
        }
    }

    const int wave = (int)((blockIdx.x * blockDim.x + threadIdx.x) >> 5);
#pragma unroll 1
    for (int tt = 0; tt < TILES_PER_WAVE; ++tt) {
        const int tile = wave * TILES_PER_WAVE + tt;
        if (tile >= n_tiles) break;
        const int b = tile * 16 + m;                // batch element = column

        const float4 xv = reinterpret_cast<const float4*>(x)[b];
        float c0,s0,c1,s1,c2,s2,c3,s3;
        __sincosf(0.5f*xv.x, &s0, &c0);
        __sincosf(0.5f*xv.y, &s1, &c1);
        __sincosf(0.5f*xv.z, &s2, &c2);
        __sincosf(0.5f*xv.w, &s3, &c3);

        float sr[8], si[8], trr[8], tri[8];

        // ---- Layer 0: RX layer on |0000> is a product state with amplitudes
        //      (-i)^popcount * (product of c/s). Each row is purely real or
        //      purely imaginary, so both halves' values are 2 muls + selects
        //      against literal zero (hand-folded: strict FP blocks 0*x folding).
        {
            const float cc = c1*c2, cs = c1*s2, sc = s1*c2, ss = s1*s2;
            const float P[8] = { cc*c3, cc*s3, cs*c3, cs*s3,
                                 sc*c3, sc*s3, ss*c3, ss*s3 };
#pragma unroll
            for (int r = 0; r < 8; ++r) {
                const int  k      = ((r>>2)&1) + ((r>>1)&1) + (r&1);
                const bool isReal = ((k & 1) == 0);          // (-i)^k real?
                const float sgn   = (k == 1 || k == 2) ? -1.f : 1.f;
                const float a  = sgn * (c0 * P[r]);          // c0-scaled component
                const float bb = sgn * (s0 * P[r]);          // s0-scaled component
                if (isReal) {       // u = {sgn*P, 0}
                    sr[r]  = lo ?  a  : 0.f;
                    si[r]  = lo ? 0.f : -bb;
                    trr[r] = lo ? 0.f :  a;
                    tri[r] = lo ? -bb : 0.f;
                } else {            // u = {0, sgn*P}
                    sr[r]  = lo ? 0.f :  bb;
                    si[r]  = lo ?  a  : 0.f;
                    trr[r] = lo ?  bb : 0.f;
                    tri[r] = lo ? 0.f :  a;
                }
            }
        }
        layer_matmul(ArF[0], AiF[0], sr, si, trr, tri, lo);

        // ---- Layers 1,2: live-state RX (one fused half-swap) + WMMA unitary
        rx_layer_live(sr, si, trr, tri, c0,s0,c1,s1,c2,s2,c3,s3);
        layer_matmul(ArF[1], AiF[1], sr, si, trr, tri, lo);

        rx_layer_live(sr, si, trr, tri, c0,s0,c1,s1,c2,s2,c3,s3);
        layer_matmul(ArF[2], AiF[2], sr, si, trr, tri, lo);

        // ---- <Z_q>: probabilities, signed row sums, half-wave reduction ----
        float p[8];
#pragma unroll
        for (int r = 0; r < 8; ++r)
            p[r] = __builtin_fmaf(sr[r], sr[r], si[r]*si[r]);
        float e0 = 0.f, e1 = 0.f, e2 = 0.f, e3 = 0.f;
#pragma unroll
        for (int r = 0; r < 8; ++r) {
            e0 += p[r];
            e1 += (r & 4) ? -p[r] : p[r];   // qubit1 -> row bit2
            e2 += (r & 2) ? -p[r] : p[r];   // qubit2 -> row bit1
            e3 += (r & 1) ? -p[r] : p[r];   // qubit3 -> row bit0
        }
        if (!lo) e0 = -e0;                  // qubit0 -> row bit3 (half-wave)
        e0 += xor16(e0);
        e1 += xor16(e1);
        e2 += xor16(e2);
        e3 += xor16(e3);
        if (lo) {
            reinterpret_cast<float4*>(out)[b] = make_float4(e0, e1, e2, e3);
        }
    }
}

extern "C" void kernel_launch(void* const* d_in, const int* in_sizes, int n_in,
                              void* d_out, int out_size, void* d_ws, size_t ws_size,
                              hipStream_t stream) {
    const float* x = (const float*)d_in[0];       // [B,4] f32
    const float* w = (const float*)d_in[1];       // [3,4,3] f32
    float* out = (float*)d_out;                   // [B,4] f32
    const int batch   = in_sizes[0] / 4;
    const int n_tiles = batch / 16;               // 16 batch elements per wave-tile
    const int tiles_per_block = WAVES_PER_BLOCK * TILES_PER_WAVE;
    const int blocks = (n_tiles + tiles_per_block - 1) / tiles_per_block;
    qfe_wmma_kernel<<<blocks, WAVES_PER_BLOCK * 32, 0, stream>>>(x, w, out, n_tiles);
    (void)n_in; (void)out_size; (void)d_ws; (void)ws_size;
}